// DualGNN_Fusion_36240934043745
// MI455X (gfx1250) — compile-verified
//
#include <hip/hip_runtime.h>
#include <hip/hip_bf16.h>

// ---------------------------------------------------------------------------
// CDNA5 (gfx1250) implementation of the dual-mesh FeaSt GNN.
// Dense transforms -> V_WMMA_F32_16X16X4_F32 (fp32 matrix path, wave32).
// Irregular graph parts -> per-edge wave kernels + float atomics.
// ---------------------------------------------------------------------------

typedef __attribute__((ext_vector_type(2))) float v2f;
typedef __attribute__((ext_vector_type(8))) float v8f;

#define LEAKY_SLOPE 0.2f

// ---------------------------------------------------------------------------
// WMMA GEMM:  Y[N,M] = act( concat(A1[N,K1], A2[rowmap(N),K2]) @ W[K,M]
//                           + bias[M] + residual[N*rstride + col] )
// One wave computes a 16x(16*NJT) strip of Y; NJT is a compile-time constant
// so the tile loop is straight-line code: NO exec masking anywhere near WMMA
// (EXEC must be all-ones for V_WMMA). Host splits each GEMM into a full-strip
// launch (NJT=4) plus at most one remainder launch (NJT=1..3).
// K is stepped by 4 via V_WMMA_F32_16X16X4_F32:
//   A layout (16x4 f32): lanes 0-15 supply K pair {0,1}, lanes 16-31 {2,3}
//   B layout (4x16 f32): mirrored; C/D: VGPR i -> rows i / i+8.
// K loop is segmented so hot loops carry no predication at all:
//   seg1 aligned pure-A1 | one branch-free boundary chunk iff K1%4 |
//   seg2 aligned pure-A2 (concat region) | one branch-free tail iff K%4.
// Wave->tile map: row tile ti is the fast index, so the 8 waves of a block
// share identical W rows (same cache lines) and touch contiguous A rows.
// ---------------------------------------------------------------------------
template <int NJT>
__global__ __launch_bounds__(256) void gemm_wmma_kernel(
    const float* __restrict__ A1, int K1,
    const float* __restrict__ A2, int K2, const int* __restrict__ ridx2,
    const float* __restrict__ W, const float* __restrict__ bias,
    const float* __restrict__ residual, int res_stride,
    float* __restrict__ Y, int N, int M, int act,
    int stripBase, int nStrips)
{
    const int tilesN = (N + 15) >> 4;
    const int tile = blockIdx.x * (blockDim.x >> 5) + (threadIdx.x >> 5);
    if (tile >= tilesN * nStrips) return;      // whole-wave exit only

    const int strip = tile / tilesN;
    const int ti    = tile - strip * tilesN;
    const int colBase = (stripBase + strip) * 64;

    const int lane = threadIdx.x & 31;
    const int half = lane >> 4;                // 0 -> K{0,1}, 1 -> K{2,3}
    const int l16  = lane & 15;
    const int K    = K1 + K2;
    const int h2   = 2 * half;

    const int rowN = ti * 16 + l16;            // A row this lane feeds
    const int rA   = (rowN < N) ? rowN : (N - 1);

    int col[NJT], cB[NJT];
#pragma unroll
    for (int j = 0; j < NJT; ++j) {
        col[j] = colBase + j * 16 + l16;       // B/C column this lane feeds
        cB[j]  = (col[j] < M) ? col[j] : (M - 1);
    }

    const float* a1row = A1 + (size_t)rA * K1;
    const float* a2row = nullptr;
    if (A2) {
        const int r2 = ridx2 ? ridx2[rA] : rA;
        a2row = A2 + (size_t)r2 * K2;
    }

    v8f acc[NJT];
#pragma unroll
    for (int j = 0; j < NJT; ++j) acc[j] = (v8f){};

    int kpos = 0;

    // ---- seg1: pure A1, aligned, unpredicated ----
#pragma unroll 2
    for (; kpos + 4 <= K1; kpos += 4) {
        const v2f a = *(const v2f*)(a1row + kpos + h2);
        const float* brow = W + (size_t)(kpos + h2) * M;
#pragma unroll
        for (int j = 0; j < NJT; ++j) {
            v2f b;
            b.x = brow[cB[j]];
            b.y = brow[(size_t)M + cB[j]];
            acc[j] = __builtin_amdgcn_wmma_f32_16x16x4_f32(
                         false, a, false, b, (short)0, acc[j], false, false);
        }
    }

    // ---- boundary chunk iff K1%4 (input layer K=6); branch-free selects ----
    if ((K1 & 3) && kpos < K) {
        const int k0 = kpos + h2;
        // unconditional clamped loads, then value selects (no divergent CF)
        const float t10 = a1row[(k0     < K1) ? k0     : (K1 - 1)];
        const float t11 = a1row[(k0 + 1 < K1) ? k0 + 1 : (K1 - 1)];
        float t20 = 0.0f, t21 = 0.0f;
        if (K2 > 0) {   // wave-uniform
            int o0 = k0 - K1;     o0 = o0 < 0 ? 0 : (o0 >= K2 ? K2 - 1 : o0);
            int o1 = k0 + 1 - K1; o1 = o1 < 0 ? 0 : (o1 >= K2 ? K2 - 1 : o1);
            t20 = a2row[o0];
            t21 = a2row[o1];
        }
        v2f a;
        a.x = (k0     < K1) ? t10 : ((k0     < K) ? t20 : 0.0f);
        a.y = (k0 + 1 < K1) ? t11 : ((k0 + 1 < K) ? t21 : 0.0f);
        const int kc0 = (k0     < K) ? k0     : (K - 1);
        const int kc1 = (k0 + 1 < K) ? k0 + 1 : (K - 1);
#pragma unroll
        for (int j = 0; j < NJT; ++j) {
            const float w0 = W[(size_t)kc0 * M + cB[j]];
            const float w1 = W[(size_t)kc1 * M + cB[j]];
            v2f b;
            b.x = (k0     < K) ? w0 : 0.0f;
            b.y = (k0 + 1 < K) ? w1 : 0.0f;
            acc[j] = __builtin_amdgcn_wmma_f32_16x16x4_f32(
                         false, a, false, b, (short)0, acc[j], false, false);
        }
        kpos += 4;
    }

    // ---- seg2: pure A2 (concat region), aligned, unpredicated ----
#pragma unroll 2
    for (; kpos + 4 <= K; kpos += 4) {
        const int o2 = kpos - K1 + h2;
        v2f a;
        a.x = a2row[o2];
        a.y = a2row[o2 + 1];
        const float* brow = W + (size_t)(kpos + h2) * M;
#pragma unroll
        for (int j = 0; j < NJT; ++j) {
            v2f b;
            b.x = brow[cB[j]];
            b.y = brow[(size_t)M + cB[j]];
            acc[j] = __builtin_amdgcn_wmma_f32_16x16x4_f32(
                         false, a, false, b, (short)0, acc[j], false, false);
        }
    }

    // ---- tail chunk iff K%4 (k >= K1 here); branch-free selects ----
    if (kpos < K) {
        const int k0 = kpos + h2;
        int o0 = k0 - K1;     o0 = o0 >= K2 ? K2 - 1 : o0;
        int o1 = k0 + 1 - K1; o1 = o1 >= K2 ? K2 - 1 : o1;
        const float t20 = a2row[o0];
        const float t21 = a2row[o1];
        v2f a;
        a.x = (k0     < K) ? t20 : 0.0f;
        a.y = (k0 + 1 < K) ? t21 : 0.0f;
        const int kc0 = (k0     < K) ? k0     : (K - 1);
        const int kc1 = (k0 + 1 < K) ? k0 + 1 : (K - 1);
#pragma unroll
        for (int j = 0; j < NJT; ++j) {
            const float w0 = W[(size_t)kc0 * M + cB[j]];
            const float w1 = W[(size_t)kc1 * M + cB[j]];
            v2f b;
            b.x = (k0     < K) ? w0 : 0.0f;
            b.y = (k0 + 1 < K) ? w1 : 0.0f;
            acc[j] = __builtin_amdgcn_wmma_f32_16x16x4_f32(
                         false, a, false, b, (short)0, acc[j], false, false);
        }
    }

    // ---- epilogue ----
#pragma unroll
    for (int j = 0; j < NJT; ++j) {
        if (col[j] < M) {
            const float bv = bias ? bias[col[j]] : 0.0f;
#pragma unroll
            for (int i = 0; i < 8; ++i) {
                const int r = ti * 16 + i + half * 8;
                if (r < N) {
                    float v = acc[j][i] + bv;
                    if (residual) v += residual[(size_t)r * res_stride + col[j]];
                    if (act) v = (v > 0.0f) ? v : LEAKY_SLOPE * v;
                    Y[(size_t)r * M + col[j]] = v;
                }
            }
        }
    }
}

// ---------------------------------------------------------------------------
// FeaSt per-edge kernel: one wave per edge (incl. self-loops appended at end).
// q = softmax(xu[src]-xu[dst]+c); msg = sum_h q[h]*xw[src][h*O+ch]
// scatter-add into agg[dst], count into cnt[dst].
// ---------------------------------------------------------------------------
__global__ __launch_bounds__(256) void feast_edge_kernel(
    const int* __restrict__ src_idx, const int* __restrict__ dst_idx,
    int E, int N,
    const float* __restrict__ xu, const float* __restrict__ cvec,
    const float* __restrict__ xw, int O,
    float* __restrict__ agg, float* __restrict__ cnt)
{
    const int wave = blockIdx.x * (blockDim.x >> 5) + (threadIdx.x >> 5);
    const int lane = threadIdx.x & 31;
    if (wave >= E + N) return;

    int s, d;
    if (wave < E) { s = src_idx[wave]; d = dst_idx[wave]; }
    else          { s = wave - E;      d = s; }

    float q[9];
    float mx = -3.402823466e38f;
#pragma unroll
    for (int h = 0; h < 9; ++h) {
        q[h] = xu[s * 9 + h] - xu[d * 9 + h] + cvec[h];
        mx = fmaxf(mx, q[h]);
    }
    float sum = 0.0f;
#pragma unroll
    for (int h = 0; h < 9; ++h) { q[h] = __expf(q[h] - mx); sum += q[h]; }
    const float inv = 1.0f / sum;

    const float* xws = xw + (size_t)s * (9 * O);
    for (int ch = lane; ch < O; ch += 32) {
        float m = 0.0f;
#pragma unroll
        for (int h = 0; h < 9; ++h) m += q[h] * xws[h * O + ch];
        atomicAdd(&agg[(size_t)d * O + ch], m * inv);
    }
    if (lane == 0) atomicAdd(&cnt[d], 1.0f);
}

// out = act( acc/max(cnt,1) + bias )
__global__ void mean_bias_act_kernel(const float* __restrict__ acc,
                                     const float* __restrict__ cnt,
                                     const float* __restrict__ bias,
                                     float* __restrict__ out,
                                     int N, int C, int act)
{
    size_t i = (size_t)blockIdx.x * blockDim.x + threadIdx.x;
    if (i >= (size_t)N * C) return;
    const int n = (int)(i / C);
    const int c = (int)(i % C);
    float v = acc[i] / fmaxf(cnt[n], 1.0f);
    if (bias) v += bias[c];
    if (act) v = (v > 0.0f) ? v : LEAKY_SLOPE * v;
    out[i] = v;
}

__global__ void fill_kernel(float* __restrict__ p, float v, size_t n)
{
    size_t i = (size_t)blockIdx.x * blockDim.x + threadIdx.x;
    if (i < n) p[i] = v;
}

// float atomic max via signed-max / unsigned-min bit trick
__device__ __forceinline__ void atomicMaxF(float* addr, float val)
{
    if (val >= 0.0f) atomicMax((int*)addr, __float_as_int(val));
    else             atomicMin((unsigned int*)addr, __float_as_uint(val));
}

__global__ void scatter_max_kernel(const float* __restrict__ x,
                                   const int* __restrict__ cluster,
                                   float* __restrict__ y, int Nin, int C)
{
    size_t i = (size_t)blockIdx.x * blockDim.x + threadIdx.x;
    if (i >= (size_t)Nin * C) return;
    const int n = (int)(i / C);
    const int c = (int)(i % C);
    atomicMaxF(&y[(size_t)cluster[n] * C + c], x[i]);
}

__global__ void neginf_zero_kernel(float* __restrict__ y, size_t n)
{
    size_t i = (size_t)blockIdx.x * blockDim.x + threadIdx.x;
    if (i < n) {
        const float v = y[i];
        y[i] = (isinf(v) && v < 0.0f) ? 0.0f : v;
    }
}

// v2f[f] = mean over 3 vertices of hv
__global__ void face_gather_mean_kernel(const float* __restrict__ hv,
                                        const int* __restrict__ fv,
                                        float* __restrict__ out, int NF, int C)
{
    size_t i = (size_t)blockIdx.x * blockDim.x + threadIdx.x;
    if (i >= (size_t)NF * C) return;
    const int f = (int)(i / C);
    const int c = (int)(i % C);
    const int* t = fv + 3 * f;
    out[i] = (hv[(size_t)t[0] * C + c] +
              hv[(size_t)t[1] * C + c] +
              hv[(size_t)t[2] * C + c]) * (1.0f / 3.0f);
}

// s[v] += hf[f] for each incident vertex; cnt[v] += 1 per incidence
__global__ void face_scatter_sum_kernel(const float* __restrict__ hf,
                                        const int* __restrict__ fv,
                                        float* __restrict__ s,
                                        float* __restrict__ cnt, int NF, int C)
{
    size_t i = (size_t)blockIdx.x * blockDim.x + threadIdx.x;
    if (i >= (size_t)NF * C) return;
    const int f = (int)(i / C);
    const int c = (int)(i % C);
    const float v = hf[i];
    const int* t = fv + 3 * f;
    atomicAdd(&s[(size_t)t[0] * C + c], v);
    atomicAdd(&s[(size_t)t[1] * C + c], v);
    atomicAdd(&s[(size_t)t[2] * C + c], v);
    if (c == 0) {
        atomicAdd(&cnt[t[0]], 1.0f);
        atomicAdd(&cnt[t[1]], 1.0f);
        atomicAdd(&cnt[t[2]], 1.0f);
    }
}

__global__ void rownorm3_kernel(float* __restrict__ y, int N)
{
    int i = blockIdx.x * blockDim.x + threadIdx.x;
    if (i >= N) return;
    const float a = y[3 * i], b = y[3 * i + 1], c = y[3 * i + 2];
    const float d = fmaxf(sqrtf(a * a + b * b + c * c), 1e-12f);
    y[3 * i]     = a / d;
    y[3 * i + 1] = b / d;
    y[3 * i + 2] = c / d;
}

// ---------------------------------------------------------------------------
struct LinP   { const float* w; const float* b; };
struct FeastP { const float* w; const float* u; const float* c; const float* bias; };
struct FusP   { const float* wv; const float* bv; const float* wf; const float* bf; };

extern "C" void kernel_launch(void* const* d_in, const int* in_sizes, int n_in,
                              void* d_out, int out_size, void* d_ws, size_t ws_size,
                              hipStream_t stream)
{
    (void)out_size; (void)ws_size;
    if (n_in < 95) return;

    constexpr int NV1 = 40000, NF1 = 80000;
    constexpr int NV2 = 20000, NF2 = 40000;
    constexpr int NV3 = 10000, NF3 = 20000;

    const float* x_v = (const float*)d_in[0];
    const float* x_f = (const float*)d_in[1];
    const int* edge_v1 = (const int*)d_in[2];  const int EV1 = in_sizes[2] / 2;
    const int* edge_f1 = (const int*)d_in[3];  const int EF1 = in_sizes[3] / 2;
    const int* edge_v2 = (const int*)d_in[4];  const int EV2 = in_sizes[4] / 2;
    const int* edge_f2 = (const int*)d_in[5];  const int EF2 = in_sizes[5] / 2;
    const int* edge_v3 = (const int*)d_in[6];  const int EV3 = in_sizes[6] / 2;
    const int* edge_f3 = (const int*)d_in[7];  const int EF3 = in_sizes[7] / 2;
    const int* cluster_v1 = (const int*)d_in[8];
    const int* cluster_f1 = (const int*)d_in[9];
    const int* cluster_v2 = (const int*)d_in[10];
    const int* cluster_f2 = (const int*)d_in[11];
    const int* fv1 = (const int*)d_in[12];
    const int* fv2 = (const int*)d_in[13];
    const int* fv3 = (const int*)d_in[14];

    // params flattened in setup_inputs() insertion order
    int pi = 15;
    auto Pp = [&]() { return (const float*)d_in[pi++]; };
    LinP   lin_in_v{Pp(), Pp()}, lin_in_f{Pp(), Pp()};
    FeastP conv_l1_v{Pp(), Pp(), Pp(), Pp()}, conv_l1_f{Pp(), Pp(), Pp(), Pp()};
    FeastP conv_l2_v{Pp(), Pp(), Pp(), Pp()}, conv_l2_f{Pp(), Pp(), Pp(), Pp()};
    FeastP conv_l3_v{Pp(), Pp(), Pp(), Pp()}, conv_l3_f{Pp(), Pp(), Pp(), Pp()};
    FeastP conv_r3_v{Pp(), Pp(), Pp(), Pp()}, conv_r3_f{Pp(), Pp(), Pp(), Pp()};
    FeastP conv_r2_v{Pp(), Pp(), Pp(), Pp()}, conv_r2_f{Pp(), Pp(), Pp(), Pp()};
    FeastP conv_r1_v{Pp(), Pp(), Pp(), Pp()}, conv_r1_f{Pp(), Pp(), Pp(), Pp()};
    FusP   fusion3{Pp(), Pp(), Pp(), Pp()};
    FusP   fusion2{Pp(), Pp(), Pp(), Pp()};
    FusP   fusion1{Pp(), Pp(), Pp(), Pp()};
    FeastP conv_out_v{Pp(), Pp(), Pp(), Pp()}, conv_out_f{Pp(), Pp(), Pp(), Pp()};
    LinP   lin_out1_v{Pp(), Pp()}, lin_out1_f{Pp(), Pp()};
    LinP   lin_out_v{Pp(), Pp()},  lin_out_f{Pp(), Pp()};

    // ---- workspace bump allocator ----
    char* wsb = (char*)d_ws;
    size_t off = 0;
    auto alloc = [&](size_t nfl) -> float* {
        float* p = (float*)(wsb + off);
        off += ((nfl * sizeof(float)) + 255) & ~(size_t)255;
        return p;
    };

    float* hin_v = alloc((size_t)NV1 * 32); float* hin_f = alloc((size_t)NF1 * 32);
    float* hv1   = alloc((size_t)NV1 * 32); float* hf1   = alloc((size_t)NF1 * 32);
    float* hv2p  = alloc((size_t)NV2 * 32); float* hf2p  = alloc((size_t)NF2 * 32);
    float* hv2   = alloc((size_t)NV2 * 64); float* hf2   = alloc((size_t)NF2 * 64);
    float* hv3p  = alloc((size_t)NV3 * 64); float* hf3p  = alloc((size_t)NF3 * 64);
    float* hv3a  = alloc((size_t)NV3 * 128); float* hf3a = alloc((size_t)NF3 * 128);
    float* hv3b  = alloc((size_t)NV3 * 64); float* hf3b  = alloc((size_t)NF3 * 64);
    float* hv3c  = alloc((size_t)NV3 * 64); float* hf3c  = alloc((size_t)NF3 * 64);
    float* hv2b  = alloc((size_t)NV2 * 32); float* hf2b  = alloc((size_t)NF2 * 32);
    float* hv2c  = alloc((size_t)NV2 * 32); float* hf2c  = alloc((size_t)NF2 * 32);
    float* hv1b  = alloc((size_t)NV1 * 32); float* hf1b  = alloc((size_t)NF1 * 32);
    float* hv1c  = alloc((size_t)NV1 * 32); float* hf1c  = alloc((size_t)NF1 * 32);
    float* hv1d  = alloc((size_t)NV1 * 32); float* hf1d  = alloc((size_t)NF1 * 32);
    float* gvb   = alloc((size_t)NV1 * 512); float* gfb  = alloc((size_t)NF1 * 512);
    // shared conv/fusion scratch (sized to maxima across all layers)
    float* xw_s  = alloc((size_t)80000 * 288);  // max N * 9*O
    float* xu_s  = alloc((size_t)80000 * 9);
    float* agg_s = alloc((size_t)80000 * 32);   // max N*O
    float* cnt_s = alloc((size_t)80000);
    float* v2f_s = alloc((size_t)80000 * 32);   // max NF*C
    float* f2v_s = alloc((size_t)40000 * 32);   // max NV*C
    float* fcnt_s = alloc((size_t)40000);

    // ---- launch helpers ----
    auto gemm = [&](const float* A1, int K1, const float* A2, int K2,
                    const int* ridx2, const float* W, const float* bias,
                    const float* res, int rstride, float* Y, int N, int M, int act) {
        const int tiles16 = (M + 15) / 16;
        const int full    = tiles16 / 4;
        const int rem     = tiles16 & 3;
        const int tilesN  = (N + 15) / 16;
        if (full) {
            long tiles = (long)tilesN * full;
            gemm_wmma_kernel<4><<<(int)((tiles + 7) / 8), 256, 0, stream>>>(
                A1, K1, A2, K2, ridx2, W, bias, res, rstride, Y, N, M, act, 0, full);
        }
        if (rem) {
            const int blocks = (tilesN + 7) / 8;
            switch (rem) {
            case 1:
                gemm_wmma_kernel<1><<<blocks, 256, 0, stream>>>(
                    A1, K1, A2, K2, ridx2, W, bias, res, rstride, Y, N, M, act, full, 1);
                break;
            case 2:
                gemm_wmma_kernel<2><<<blocks, 256, 0, stream>>>(
                    A1, K1, A2, K2, ridx2, W, bias, res, rstride, Y, N, M, act, full, 1);
                break;
            default:
                gemm_wmma_kernel<3><<<blocks, 256, 0, stream>>>(
                    A1, K1, A2, K2, ridx2, W, bias, res, rstride, Y, N, M, act, full, 1);
                break;
            }
        }
    };
    auto fill = [&](float* p, float v, size_t n) {
        fill_kernel<<<(int)((n + 255) / 256), 256, 0, stream>>>(p, v, n);
    };
    auto run_feast = [&](const float* X1, int K1, const float* X2, int K2,
                         const int* ridx2, const int* edge, int E, int N, int O,
                         const FeastP& Q, float* out) {
        gemm(X1, K1, X2, K2, ridx2, Q.w, nullptr, nullptr, 0, xw_s, N, 9 * O, 0);
        gemm(X1, K1, X2, K2, ridx2, Q.u, nullptr, nullptr, 0, xu_s, N, 9, 0);
        fill(agg_s, 0.0f, (size_t)N * O);
        fill(cnt_s, 0.0f, (size_t)N);
        const int waves = E + N;
        feast_edge_kernel<<<(waves + 7) / 8, 256, 0, stream>>>(
            edge, edge + E, E, N, xu_s, Q.c, xw_s, O, agg_s, cnt_s);
        mean_bias_act_kernel<<<(int)(((size_t)N * O + 255) / 256), 256, 0, stream>>>(
            agg_s, cnt_s, Q.bias, out, N, O, 1);
    };
    auto run_pool = [&](const float* x, const int* cl, int Nin, int Nout, int C,
                        float* y) {
        fill(y, -__builtin_inff(), (size_t)Nout * C);
        scatter_max_kernel<<<(int)(((size_t)Nin * C + 255) / 256), 256, 0, stream>>>(
            x, cl, y, Nin, C);
        neginf_zero_kernel<<<(int)(((size_t)Nout * C + 255) / 256), 256, 0, stream>>>(
            y, (size_t)Nout * C);
    };
    auto run_fusion = [&](const float* hv, const float* hf, const int* fv,
                          int NVn, int NFn, int C, const FusP& Q,
                          float* outv, float* outf) {
        face_gather_mean_kernel<<<(int)(((size_t)NFn * C + 255) / 256), 256, 0, stream>>>(
            hv, fv, v2f_s, NFn, C);
        fill(f2v_s, 0.0f, (size_t)NVn * C);
        fill(fcnt_s, 0.0f, (size_t)NVn);
        face_scatter_sum_kernel<<<(int)(((size_t)NFn * C + 255) / 256), 256, 0, stream>>>(
            hf, fv, f2v_s, fcnt_s, NFn, C);
        mean_bias_act_kernel<<<(int)(((size_t)NVn * C + 255) / 256), 256, 0, stream>>>(
            f2v_s, fcnt_s, nullptr, f2v_s, NVn, C, 0);
        gemm(hv, C, f2v_s, C, nullptr, Q.wv, Q.bv, nullptr, 0, outv, NVn, C, 1);
        gemm(hf, C, v2f_s, C, nullptr, Q.wf, Q.bf, nullptr, 0, outf, NFn, C, 1);
    };

    // ---- forward pass ----
    gemm(x_v, 6, nullptr, 0, nullptr, lin_in_v.w, lin_in_v.b, nullptr, 0, hin_v, NV1, 32, 1);
    gemm(x_f, 6, nullptr, 0, nullptr, lin_in_f.w, lin_in_f.b, nullptr, 0, hin_f, NF1, 32, 1);

    run_feast(hin_v, 32, nullptr, 0, nullptr, edge_v1, EV1, NV1, 32, conv_l1_v, hv1);
    run_feast(hin_f, 32, nullptr, 0, nullptr, edge_f1, EF1, NF1, 32, conv_l1_f, hf1);

    run_pool(hv1, cluster_v1, NV1, NV2, 32, hv2p);
    run_pool(hf1, cluster_f1, NF1, NF2, 32, hf2p);
    run_feast(hv2p, 32, nullptr, 0, nullptr, edge_v2, EV2, NV2, 64, conv_l2_v, hv2);
    run_feast(hf2p, 32, nullptr, 0, nullptr, edge_f2, EF2, NF2, 64, conv_l2_f, hf2);

    run_pool(hv2, cluster_v2, NV2, NV3, 64, hv3p);
    run_pool(hf2, cluster_f2, NF2, NF3, 64, hf3p);
    run_feast(hv3p, 64, nullptr, 0, nullptr, edge_v3, EV3, NV3, 128, conv_l3_v, hv3a);
    run_feast(hf3p, 64, nullptr, 0, nullptr, edge_f3, EF3, NF3, 128, conv_l3_f, hf3a);
    run_feast(hv3a, 128, nullptr, 0, nullptr, edge_v3, EV3, NV3, 64, conv_r3_v, hv3b);
    run_feast(hf3a, 128, nullptr, 0, nullptr, edge_f3, EF3, NF3, 64, conv_r3_f, hf3b);
    run_fusion(hv3b, hf3b, fv3, NV3, NF3, 64, fusion3, hv3c, hf3c);

    // conv_r2 on concat([hv2, hv3c[cluster_v2]]) without materializing the concat
    run_feast(hv2, 64, hv3c, 64, cluster_v2, edge_v2, EV2, NV2, 32, conv_r2_v, hv2b);
    run_feast(hf2, 64, hf3c, 64, cluster_f2, edge_f2, EF2, NF2, 32, conv_r2_f, hf2b);
    run_fusion(hv2b, hf2b, fv2, NV2, NF2, 32, fusion2, hv2c, hf2c);

    run_feast(hv1, 32, hv2c, 32, cluster_v1, edge_v1, EV1, NV1, 32, conv_r1_v, hv1b);
    run_feast(hf1, 32, hf2c, 32, cluster_f1, edge_f1, EF1, NF1, 32, conv_r1_f, hf1b);
    run_fusion(hv1b, hf1b, fv1, NV1, NF1, 32, fusion1, hv1c, hf1c);

    run_feast(hv1c, 32, nullptr, 0, nullptr, edge_v1, EV1, NV1, 32, conv_out_v, hv1d);
    run_feast(hf1c, 32, nullptr, 0, nullptr, edge_f1, EF1, NF1, 32, conv_out_f, hf1d);

    gemm(hv1d, 32, nullptr, 0, nullptr, lin_out1_v.w, lin_out1_v.b, nullptr, 0, gvb, NV1, 512, 1);
    gemm(hf1d, 32, nullptr, 0, nullptr, lin_out1_f.w, lin_out1_f.b, nullptr, 0, gfb, NF1, 512, 1);

    float* feat_v = (float*)d_out;
    float* feat_f = (float*)d_out + (size_t)NV1 * 3;
    // residuals: xyz = x_v[:, 0:3] (stride 6); nf = x_f[:, 3:6] (stride 6)
    gemm(gvb, 512, nullptr, 0, nullptr, lin_out_v.w, lin_out_v.b, x_v,     6, feat_v, NV1, 3, 0);
    gemm(gfb, 512, nullptr, 0, nullptr, lin_out_f.w, lin_out_f.b, x_f + 3, 6, feat_f, NF1, 3, 0);

    rownorm3_kernel<<<(NF1 + 255) / 256, 256, 0, stream>>>(feat_f, NF1);
}